// VecInt_43765716747058
// MI455X (gfx1250) — compile-verified
//
#include <hip/hip_runtime.h>

// VecInt (scaling & squaring diffeomorphic integration), shape (B,160,192,160,3) fp32.
// disp = vel / 2^7 ; 7x: disp = disp + trilinear(disp, grid + disp)
//
// MI455X strategy: per-batch ping-pong so the 118 MB (in+out) working set stays
// resident in the 192 MB L2 across all 7 squaring steps. Gather-bound, wave32,
// fp32 throughout (matches reference numerics). No matrix ops exist in this
// problem -> no WMMA by design.

#define DD 160
#define HH 192
#define WW 160
#define NVOX (DD * HH * WW)

struct F3 { float z, y, x; };   // channel order: [0]=D(z), [1]=H(y), [2]=W(x)

__device__ __forceinline__ void gather_acc(const float* __restrict__ f, int idx,
                                           float w, float& az, float& ay, float& ax) {
    const F3 v = *(reinterpret_cast<const F3*>(f) + idx);   // 12B packed load
    az = fmaf(w, v.z, az);
    ay = fmaf(w, v.y, ay);
    ax = fmaf(w, v.x, ax);
}

__global__ __launch_bounds__(256)
void vecint_init(const float4* __restrict__ v, float4* __restrict__ o, int n4) {
    int i = blockIdx.x * 256 + threadIdx.x;
    if (i < n4) {
        const float s = 1.0f / 128.0f;   // 2^-INT_STEPS, INT_STEPS=7
        float4 a = v[i];
        o[i] = make_float4(a.x * s, a.y * s, a.z * s, a.w * s);
    }
}

__global__ __launch_bounds__(256)
void vecint_step(const float* __restrict__ din, float* __restrict__ dout) {
    int t = blockIdx.x * 256 + threadIdx.x;
    if (t >= NVOX) return;

    int x = t % WW;
    int r = t / WW;
    int y = r % HH;
    int z = r / HH;

    const F3 dv = *(reinterpret_cast<const F3*>(din) + t);  // coalesced stream read

    // sample location = grid + disp, clipped to volume extent (matches jnp.clip)
    float lz = fminf(fmaxf((float)z + dv.z, 0.0f), (float)(DD - 1));
    float ly = fminf(fmaxf((float)y + dv.y, 0.0f), (float)(HH - 1));
    float lx = fminf(fmaxf((float)x + dv.x, 0.0f), (float)(WW - 1));

    float fz = floorf(lz), fy = floorf(ly), fx = floorf(lx);
    int z0 = (int)fz, y0 = (int)fy, x0 = (int)fx;
    int z1 = min(z0 + 1, DD - 1);
    int y1 = min(y0 + 1, HH - 1);
    int x1 = min(x0 + 1, WW - 1);

    float tz = lz - fz, ty = ly - fy, tx = lx - fx;
    float sz = 1.0f - tz, sy = 1.0f - ty, sx = 1.0f - tx;

    int bz0 = z0 * (HH * WW), bz1 = z1 * (HH * WW);
    int by0 = y0 * WW,        by1 = y1 * WW;

    float wzy00 = sz * sy, wzy01 = sz * ty, wzy10 = tz * sy, wzy11 = tz * ty;

    float az = 0.0f, ay = 0.0f, ax = 0.0f;
    gather_acc(din, bz0 + by0 + x0, wzy00 * sx, az, ay, ax);   // c000
    gather_acc(din, bz0 + by0 + x1, wzy00 * tx, az, ay, ax);   // c001
    gather_acc(din, bz0 + by1 + x0, wzy01 * sx, az, ay, ax);   // c010
    gather_acc(din, bz0 + by1 + x1, wzy01 * tx, az, ay, ax);   // c011
    gather_acc(din, bz1 + by0 + x0, wzy10 * sx, az, ay, ax);   // c100
    gather_acc(din, bz1 + by0 + x1, wzy10 * tx, az, ay, ax);   // c101
    gather_acc(din, bz1 + by1 + x0, wzy11 * sx, az, ay, ax);   // c110
    gather_acc(din, bz1 + by1 + x1, wzy11 * tx, az, ay, ax);   // c111

    F3 o;
    o.z = dv.z + az;
    o.y = dv.y + ay;
    o.x = dv.x + ax;
    *(reinterpret_cast<F3*>(dout) + t) = o;                    // coalesced stream write
}

extern "C" void kernel_launch(void* const* d_in, const int* in_sizes, int n_in,
                              void* d_out, int out_size, void* d_ws, size_t ws_size,
                              hipStream_t stream) {
    (void)n_in; (void)out_size; (void)ws_size;

    const size_t NF = (size_t)NVOX * 3;                 // floats per batch (14,745,600)
    const int nb = (int)((size_t)in_sizes[0] / NF);     // batch count (2)

    const float* vel = (const float*)d_in[0];
    float* out = (float*)d_out;
    float* ws  = (float*)d_ws;                          // one batch-sized ping buffer (~59 MB)

    const int n4 = (int)(NF / 4);                       // NF divisible by 4
    dim3 blk(256);
    dim3 grid_init((n4 + 255) / 256);
    dim3 grid_step((NVOX + 255) / 256);

    // Per-batch sequential: keeps in+out (118 MB) inside the 192 MB L2 for all
    // 7 squaring steps. 7 is odd, so start in ws -> final step writes d_out.
    for (int b = 0; b < nb; ++b) {
        const float* vb = vel + (size_t)b * NF;
        float* A = ws;                       // ping (scratch)
        float* B = out + (size_t)b * NF;     // pong (holds final result)

        vecint_init<<<grid_init, blk, 0, stream>>>((const float4*)vb, (float4*)A, n4);

        float* src = A;
        float* dst = B;
        for (int s = 0; s < 7; ++s) {
            vecint_step<<<grid_step, blk, 0, stream>>>(src, dst);
            float* tmp = src; src = dst; dst = tmp;
        }
        // last write (step 7) landed in B == d_out slice for this batch
    }
}